// HyperbolicCrossAttention_40243843564011
// MI455X (gfx1250) — compile-verified
//
#include <hip/hip_runtime.h>
#include <hip/hip_bf16.h>

typedef __attribute__((ext_vector_type(2))) float v2f;
typedef __attribute__((ext_vector_type(8))) float v8f;

#define BD     (128 * 512)
#define DDIM   512
#define MAXN   (1.0f - 4e-3f)   // (1 - BALL_EPS)/sqrt(c)
#define MINN   1e-15f
#define AEPS   1e-5f
#define LOG2E  1.4426950408889634f
#define LN2    0.6931471805599453f

__device__ __forceinline__ float fast_rcp(float x) { return __builtin_amdgcn_rcpf(x); }

__device__ __forceinline__ float artanh_f(float x) {
    // clip to [-1+1e-7, 1-1e-7] like the reference; 0.5*ln((1+x)/(1-x)) via v_log/v_rcp
    x = fminf(fmaxf(x, -0.9999999f), 0.9999999f);
    return (0.5f * LN2) * __builtin_amdgcn_logf((1.0f + x) * fast_rcp(1.0f - x));
}

__device__ __forceinline__ float tanh_f(float z) {
    z = fminf(fmaxf(z, -15.0f), 15.0f);
    float e = __builtin_amdgcn_exp2f((2.0f * LOG2E) * z);   // e^{2z}
    return (e - 1.0f) * fast_rcp(e + 1.0f);
}

// scalar Poincare projx == clamp to +/-MAX_NORM
__device__ __forceinline__ float project_f(float r) {
    return fminf(fmaxf(r, -MAXN), MAXN);
}

// ---------------------------------------------------------------------------
// mx = X @ W^T via V_WMMA_F32_16X16X4_F32, fused mobius_matvec epilogue.
// One wave computes one 16x16 output tile; 4 waves per block cover 16x64.
// If do_log != 0, store artanh(project(res))  (== logmap0 of the projected V).
// ---------------------------------------------------------------------------
__global__ __launch_bounds__(128) void mobius_matvec_wmma(
    const float* __restrict__ X,   // [128, 512] row-major
    const float* __restrict__ W,   // [512, 512] row-major; mx[b,n] = sum_k X[b,k]*W[n,k]
    float* __restrict__ Out,       // [128, 512]
    int do_log)
{
    const int D    = DDIM;
    const int lane = threadIdx.x & 31;
    const int wave = threadIdx.x >> 5;            // 0..3
    const int m0   = blockIdx.y * 16;             // batch tile
    const int n0   = blockIdx.x * 64 + wave * 16; // output-feature tile

    const int l    = lane & 15;
    const int half = lane >> 4;                   // selects K pair (0,1) vs (2,3)

    // A frag: lane -> row m = m0+l, K base offset 2*half  (64-bit contiguous load)
    const float* arow = X + (m0 + l) * D;
    // B frag: B[k][n] = W[n][k] -> lane reads row n = n0+l of W along k
    const float* brow = W + (n0 + l) * D;

    v8f c = {};
    #pragma unroll 8
    for (int k0 = 0; k0 < DDIM; k0 += 4) {
        const int kb = k0 + 2 * half;
        v2f a, b;
        a.x = arow[kb];  a.y = arow[kb + 1];
        b.x = brow[kb];  b.y = brow[kb + 1];
        c = __builtin_amdgcn_wmma_f32_16x16x4_f32(
                /*neg_a=*/false, a, /*neg_b=*/false, b,
                /*c_mod=*/(short)0, c, /*reuse_a=*/false, /*reuse_b=*/false);
    }

    // Epilogue: lane holds column n = n0 + l, rows m = m0 + 8*half + r.
    const int n = n0 + l;
    const int mbase = m0 + 8 * half;
    #pragma unroll
    for (int r = 0; r < 8; ++r) {
        const int m = mbase + r;
        float mx  = c[r];
        float xv  = X[m * D + n];                 // x at the same [b, i] position
        float xa  = fmaxf(fabsf(xv), MINN);
        float mxa = fmaxf(fabsf(mx), MINN);
        // res = tanh(|mx|/|x| * artanh(|x|)) * sign(mx)
        float t   = tanh_f(mxa * fast_rcp(xa) * artanh_f(xa));
        float res = copysignf(t, mx);
        res = project_f(res);
        if (do_log) res = artanh_f(res);          // fused logmap0 for V
        Out[m * D + n] = res;
    }
}

// ---------------------------------------------------------------------------
// Inverse-distance attention + hyperbolic midpoint.
// One thread per (b, i); K[b,:] and logV[b,:] staged in LDS via the CDNA5
// async global->LDS path (global_load_async_to_lds_b128 + s_wait_asynccnt).
// ---------------------------------------------------------------------------
__global__ __launch_bounds__(256) void attend_kernel(
    const float* __restrict__ Q,     // [128, 512] projected query points
    const float* __restrict__ K,     // [128, 512] projected key points
    const float* __restrict__ logV,  // [128, 512] logmap0 of projected values
    float* __restrict__ Y)           // [128, 512]
{
    __shared__ float sK[DDIM];
    __shared__ float sV[DDIM];

    const int b = blockIdx.y;
    const int i = blockIdx.x * 256 + threadIdx.x;
    const int base = b * DDIM;

    // Stage K row into sK (threads 0..127) and logV row into sV (threads
    // 128..255), 16 bytes per lane, one async instruction per wave.
    {
        const int t      = threadIdx.x;
        const bool isK   = (t < 128);
        const int  e     = (isK ? t : (t - 128)) * 4;       // element offset
        const float* gsrc = (isK ? K : logV) + base + e;
        float*       ldst = (isK ? sK : sV) + e;
        unsigned lds_addr = (unsigned)(unsigned long long)ldst; // flat->LDS: low 32 bits
        asm volatile("global_load_async_to_lds_b128 %0, %1, off"
                     :: "v"(lds_addr), "v"(gsrc) : "memory");
        asm volatile("s_wait_asynccnt 0x0" ::: "memory");
    }
    __syncthreads();

    const float x  = -Q[base + i];    // mobius_add(-q, k)
    const float x2 = x * x;

    float accw = 0.0f, accv = 0.0f;
    #pragma unroll 4
    for (int j = 0; j < DDIM; ++j) {
        const float y  = sK[j];
        const float xy = x * y;
        const float y2 = y * y;
        const float num = (1.0f + 2.0f * xy + y2) * x + (1.0f - x2) * y;
        const float opxy = 1.0f + xy;
        const float den  = fmaxf(opxy * opxy, MINN);        // 1+2xy+x^2y^2 = (1+xy)^2
        const float d    = num * fast_rcp(den);
        // dist = 2*artanh(|d|); w = 1/(dist + eps)
        const float t    = fminf(fabsf(d), 0.9999999f);
        const float dist = LN2 * __builtin_amdgcn_logf((1.0f + t) * fast_rcp(1.0f - t));
        const float w    = fast_rcp(dist + AEPS);
        accw += w;
        accv += w * sV[j];
    }

    // normalized weighted midpoint in tangent space, then expmap0 + project
    const float u = accv * fast_rcp(accw);
    Y[base + i] = project_f(tanh_f(u));
}

extern "C" void kernel_launch(void* const* d_in, const int* in_sizes, int n_in,
                              void* d_out, int out_size, void* d_ws, size_t ws_size,
                              hipStream_t stream) {
    (void)in_sizes; (void)n_in; (void)out_size; (void)ws_size;
    const float* x1  = (const float*)d_in[0];
    const float* x2  = (const float*)d_in[1];
    const float* Wq1 = (const float*)d_in[2];
    const float* Wk2 = (const float*)d_in[3];
    const float* Wv2 = (const float*)d_in[4];
    const float* Wq2 = (const float*)d_in[5];
    const float* Wk1 = (const float*)d_in[6];
    const float* Wv1 = (const float*)d_in[7];

    float* ws  = (float*)d_ws;
    float* Q1  = ws + 0 * BD;
    float* K2  = ws + 1 * BD;
    float* LV2 = ws + 2 * BD;
    float* Q2  = ws + 3 * BD;
    float* K1  = ws + 4 * BD;
    float* LV1 = ws + 5 * BD;

    float* y1 = (float*)d_out;
    float* y2 = y1 + BD;

    dim3 ggrid(DDIM / 64, 128 / 16);   // (8, 8) blocks, 4 waves/block
    dim3 gblock(128);
    mobius_matvec_wmma<<<ggrid, gblock, 0, stream>>>(x1, Wq1, Q1,  0);
    mobius_matvec_wmma<<<ggrid, gblock, 0, stream>>>(x2, Wk2, K2,  0);
    mobius_matvec_wmma<<<ggrid, gblock, 0, stream>>>(x2, Wv2, LV2, 1);
    mobius_matvec_wmma<<<ggrid, gblock, 0, stream>>>(x2, Wq2, Q2,  0);
    mobius_matvec_wmma<<<ggrid, gblock, 0, stream>>>(x1, Wk1, K1,  0);
    mobius_matvec_wmma<<<ggrid, gblock, 0, stream>>>(x1, Wv1, LV1, 1);

    dim3 agrid(DDIM / 256, 128);       // (2, 128)
    attend_kernel<<<agrid, dim3(256), 0, stream>>>(Q1, K2, LV2, y1);
    attend_kernel<<<agrid, dim3(256), 0, stream>>>(Q2, K1, LV1, y2);
}